// GraphConvolution_584115552306
// MI455X (gfx1250) — compile-verified
//
#include <hip/hip_runtime.h>

// ---------------------------------------------------------------------------
// GraphConvolution for MI455X (gfx1250, wave32).
//   h   = x @ W              (fp32 WMMA 16x16x4, one 16x16 tile per wave)
//   out = bias + scatter_add(row, vals * h[col])   (global_atomic_add_f32)
// x:[N,128] f32, W:[128,64] f32, row/col:[E] i32, vals:[E] f32, bias:[64] f32
// ---------------------------------------------------------------------------

typedef __attribute__((ext_vector_type(2))) float v2f;
typedef __attribute__((ext_vector_type(8))) float v8f;

#define IN_F   128
#define OUT_F  64

// ---------------------------------------------------------------------------
// Kernel 0: out[n][f] = bias[f]
// ---------------------------------------------------------------------------
__global__ __launch_bounds__(256)
void gc_init_bias(float* __restrict__ out, const float* __restrict__ bias,
                  int total) {
    int gid = blockIdx.x * blockDim.x + threadIdx.x;
    if (gid < total) {
        out[gid] = bias[gid & (OUT_F - 1)];
    }
}

// ---------------------------------------------------------------------------
// Kernel 1: h = x @ W via V_WMMA_F32_16X16X4_F32.
// 256 threads = 8 waves per block; each wave computes one 16x16 tile of h.
// A-layout : lane L -> m = L%16 ; VGPR v -> k = kbase + 2*(L/16) + v
// B-layout : lane L -> n = L%16 ; VGPR v -> k = kbase + 2*(L/16) + v
// C/D      : VGPR v, lane L -> (m = v + 8*(L/16), n = L%16)
// ---------------------------------------------------------------------------
__global__ __launch_bounds__(256)
void gc_gemm_wmma(const float* __restrict__ x,      // [N, IN_F]
                  const float* __restrict__ w,      // [IN_F, OUT_F]
                  float* __restrict__ h,            // [N, OUT_F]
                  int nNodes, int colTiles, int totalTiles) {
    const int wave = threadIdx.x >> 5;          // 0..7
    const int lane = threadIdx.x & 31;          // 0..31
    const int tile = blockIdx.x * 8 + wave;
    if (tile >= totalTiles) return;             // wave-uniform exit

    const int rowTile = tile / colTiles;
    const int colTile = tile - rowTile * colTiles;

    const int m  = lane & 15;                   // 0..15
    const int hi = lane >> 4;                   // 0 or 1 (upper K half / upper M half)
    const int rowA = rowTile * 16 + m;          // A row this lane feeds
    const int n    = colTile * 16 + m;          // B/C/D column this lane feeds

    // Clamp OOB rows to row 0 (keeps EXEC all-ones for WMMA; stores are guarded).
    const int rowSafe = (rowA < nNodes) ? rowA : 0;
    const float* __restrict__ xRow = x + (size_t)rowSafe * IN_F + 2 * hi;
    const float* __restrict__ wCol = w + (size_t)(2 * hi) * OUT_F + n;

    v8f acc = {};
#pragma unroll
    for (int k = 0; k < IN_F; k += 4) {
        // A fragment: contiguous 8-byte load per lane.
        v2f a = *(const v2f*)(xRow + k);
        // B fragment: two coalesced row loads of W (hot in L2/WGP$).
        v2f b;
        b.x = wCol[(size_t)k * OUT_F];
        b.y = wCol[(size_t)(k + 1) * OUT_F];
        acc = __builtin_amdgcn_wmma_f32_16x16x4_f32(
            /*neg_a=*/false, a, /*neg_b=*/false, b,
            /*c_mod=*/(short)0, acc, /*reuse_a=*/false, /*reuse_b=*/false);
    }

    // Store the 16x16 tile: VGPR v -> row (v + 8*hi), column n.
    const int rowBase = rowTile * 16 + 8 * hi;
#pragma unroll
    for (int v = 0; v < 8; ++v) {
        const int r = rowBase + v;
        if (r < nNodes) {
            h[(size_t)r * OUT_F + n] = acc[v];
        }
    }
}

// ---------------------------------------------------------------------------
// Kernel 2: per edge e: out[row[e]][:] += vals[e] * h[col[e]][:]
// 16 lanes per edge, float4 per lane -> each half-wave gathers a coalesced
// 256B slab of h (L2-resident) and issues 4x global_atomic_add_f32.
// ---------------------------------------------------------------------------
__global__ __launch_bounds__(256)
void gc_scatter_atomic(const float* __restrict__ h,
                       const int* __restrict__ row,
                       const int* __restrict__ col,
                       const float* __restrict__ vals,
                       float* __restrict__ out,
                       int nEdges) {
    const long long gid = (long long)blockIdx.x * blockDim.x + threadIdx.x;
    const int e = (int)(gid >> 4);
    if (e >= nEdges) return;
    const int f = ((int)gid & 15) * 4;          // feature offset 0..60

    const float v = vals[e];
    const int   c = col[e];
    const int   r = row[e];

    const float4 m = *(const float4*)(h + (size_t)c * OUT_F + f);
    float* __restrict__ o = out + (size_t)r * OUT_F + f;
    atomicAdd(o + 0, v * m.x);
    atomicAdd(o + 1, v * m.y);
    atomicAdd(o + 2, v * m.z);
    atomicAdd(o + 3, v * m.w);
}

// ---------------------------------------------------------------------------
// Host launcher
// ---------------------------------------------------------------------------
extern "C" void kernel_launch(void* const* d_in, const int* in_sizes, int n_in,
                              void* d_out, int out_size, void* d_ws, size_t ws_size,
                              hipStream_t stream) {
    const float* x    = (const float*)d_in[0];
    const int*   row  = (const int*)  d_in[1];
    const int*   col  = (const int*)  d_in[2];
    const float* vals = (const float*)d_in[3];
    const float* w    = (const float*)d_in[4];
    const float* bias = (const float*)d_in[5];
    float*       out  = (float*)d_out;
    float*       h    = (float*)d_ws;           // [nNodes, OUT_F] fp32 scratch

    const int nNodes = in_sizes[0] / IN_F;
    const int nEdges = in_sizes[1];

    // 1) out = bias (broadcast)
    {
        const int total = nNodes * OUT_F;
        const int blocks = (total + 255) / 256;
        gc_init_bias<<<blocks, 256, 0, stream>>>(out, bias, total);
    }

    // 2) h = x @ W via WMMA
    {
        const int rowTiles   = (nNodes + 15) / 16;
        const int colTiles   = OUT_F / 16;       // 4
        const int totalTiles = rowTiles * colTiles;
        const int blocks     = (totalTiles + 7) / 8;   // 8 waves per block
        gc_gemm_wmma<<<blocks, 256, 0, stream>>>(x, w, h, nNodes, colTiles, totalTiles);
    }

    // 3) scatter-add with fp32 atomics (out already holds bias)
    {
        const long long threads = (long long)nEdges * 16;
        const int blocks = (int)((threads + 255) / 256);
        gc_scatter_atomic<<<blocks, 256, 0, stream>>>(h, row, col, vals, out, nEdges);
    }
}